// GaussianScalarCompander_70274254897551
// MI455X (gfx1250) — compile-verified
//
#include <hip/hip_runtime.h>
#include <math.h>

#define NQ 1024   // number of centers (reference N)

static constexpr float kSqrt6    = 2.44948974278317809820f;
static constexpr float kSqrt3    = 1.73205080756887729353f;
static constexpr float kInvSqrt6 = 0.40824829046386301637f;
static constexpr float kInvNQ    = 1.0f / (float)NQ;

// CDF_Y(v) = 0.5*erf(sqrt(3)*erfinv(2v-1)) + 0.5, with exact endpoint handling.
__device__ __forceinline__ float cdf_y(float v) {
    if (v <= 0.0f) return 0.0f;
    if (v >= 1.0f) return 1.0f;
    return fmaf(0.5f, erff(kSqrt3 * erfinvf(fmaf(2.0f, v, -1.0f))), 0.5f);
}

// ---------------------------------------------------------------------------
// Kernel 1: build the two 1024-entry lookup tables in workspace.
//   ws[0      .. NQ)  : x_hat(k)      = sqrt(6)*erfinv(2*c_k - 1)
//   ws[NQ .. 2*NQ)    : likelihood(k) = CDF_Y((k+1)/N) - CDF_Y(k/N)
// ---------------------------------------------------------------------------
__global__ void __launch_bounds__(256)
build_tables_kernel(float* __restrict__ ws) {
    const int k = blockIdx.x * blockDim.x + threadIdx.x;
    if (k >= NQ) return;
    const float c = ((float)k + 0.5f) * kInvNQ;
    ws[k] = kSqrt6 * erfinvf(fmaf(2.0f, c, -1.0f));
    const float fk = (float)k;
    ws[NQ + k] = cdf_y((fk + 1.0f) * kInvNQ) - cdf_y(fk * kInvNQ);
}

typedef __attribute__((ext_vector_type(4))) float floatx4;
typedef __attribute__((ext_vector_type(4))) int   intx4;

// ---------------------------------------------------------------------------
// Kernel 2: streaming compander. 256 threads/block, 4 elems/thread (float4).
// Stages the 8KB tables into LDS via CDNA5 async global->LDS DMA (ASYNCcnt),
// overlapped with a global_prefetch of this thread's input cacheline, then
// does the O(1) index computation (uniform centers => argmin == floor(y*N)).
// ---------------------------------------------------------------------------
__global__ void __launch_bounds__(256)
compand_kernel(const float* __restrict__ x, const float* __restrict__ tabs,
               float* __restrict__ out, int n) {
    __shared__ __align__(16) float lds_xhat[NQ];
    __shared__ __align__(16) float lds_lik[NQ];

    const int tid  = threadIdx.x;
    const int base = (blockIdx.x * 256 + tid) * 4;

    // Prefetch this thread's input so the HBM fetch overlaps table staging.
    __builtin_prefetch(x + base, 0, 3);

#if defined(__gfx1250__) && __has_builtin(__builtin_amdgcn_global_load_async_to_lds_b128)
    {
        // Async DMA: 256 lanes x 16B = 4KB per table, tracked by ASYNCcnt.
        typedef __attribute__((address_space(1))) intx4 gv4i;
        typedef __attribute__((address_space(3))) intx4 lv4i;
        gv4i* g  = (gv4i*)(uintptr_t)tabs;          // drop const; AS1 v4i
        lv4i* lx = (lv4i*)lds_xhat;                 // AS3 v4i
        lv4i* ll = (lv4i*)lds_lik;
        __builtin_amdgcn_global_load_async_to_lds_b128(g + tid,            lx + tid, 0, 0);
        __builtin_amdgcn_global_load_async_to_lds_b128(g + (NQ / 4) + tid, ll + tid, 0, 0);
#if __has_builtin(__builtin_amdgcn_s_wait_asynccnt)
        __builtin_amdgcn_s_wait_asynccnt(0);
#else
        asm volatile("s_wait_asynccnt 0" ::: "memory");
#endif
    }
#else
    // Fallback staging path (also what the host pass parses).
    for (int i = tid; i < NQ; i += 256) {
        lds_xhat[i] = tabs[i];
        lds_lik[i]  = tabs[NQ + i];
    }
#endif
    __syncthreads();

    if (base + 4 > n) return;

    const floatx4 xv = *(const floatx4*)(x + base);
    floatx4 xh, lk, yh;
#pragma unroll
    for (int i = 0; i < 4; ++i) {
        // y = 0.5*erf(x/sqrt(6)) + 0.5  in [0,1]
        const float y = fmaf(0.5f, erff(xv[i] * kInvSqrt6), 0.5f);
        // uniform centers => nearest center index is floor(y*N), clamped
        int k = (int)(y * (float)NQ);
        k = k < 0 ? 0 : k;
        k = k > (NQ - 1) ? (NQ - 1) : k;
        xh[i] = lds_xhat[k];                    // sqrt(6)*erfinv(2*y_hat-1)
        lk[i] = lds_lik[k];                     // CDF_Y((k+1)/N)-CDF_Y(k/N)
        yh[i] = ((float)k + 0.5f) * kInvNQ;     // centers[k]
    }

    // Output layout: [x_hat (n) | likelihood (n) | y_hat (n)]
    *(floatx4*)(out + base)         = xh;
    *(floatx4*)(out + n + base)     = lk;
    *(floatx4*)(out + 2 * n + base) = yh;
}

extern "C" void kernel_launch(void* const* d_in, const int* in_sizes, int n_in,
                              void* d_out, int out_size, void* d_ws, size_t ws_size,
                              hipStream_t stream) {
    const float* x  = (const float*)d_in[0];   // [B] float32
    // d_in[1] (centers) is implied by k: centers[k] = (k+0.5)/NQ — recomputed.
    float* ws  = (float*)d_ws;                 // 2*NQ floats of scratch tables
    float* out = (float*)d_out;                // [3*B] float32
    const int n = in_sizes[0];                 // B = 262144

    build_tables_kernel<<<(NQ + 255) / 256, 256, 0, stream>>>(ws);

    const int elems_per_block = 256 * 4;
    const int blocks = (n + elems_per_block - 1) / elems_per_block;
    compand_kernel<<<blocks, 256, 0, stream>>>(x, ws, out, n);
}